// DeformConvSquareDepthWise_20109036880558
// MI455X (gfx1250) — compile-verified
//
#include <hip/hip_runtime.h>
#include <hip/hip_bf16.h>

#define Hh 256
#define Ww 256
#define HWp 65536
#define Cc 64
#define Nn 8
#define STRX 68   // padded LDS stride (floats): (m*68+c)%64 = (4m+c)%64 -> conflict-free

typedef __attribute__((ext_vector_type(2))) float v2f;
typedef __attribute__((ext_vector_type(8))) float v8f;

// ---------------------------------------------------------------------------
// Kernel 1: fused pointwise GEMM (WMMA f32 16x16x4) + offset-scalar reduction
//   y[n,hw,c]  (NHWC, workspace)   s[n,hw]  (workspace)
// Block: 256 threads (8 waves), 128 pixels x 64 channels per block.
// ---------------------------------------------------------------------------
__global__ __launch_bounds__(256) void pw_gemm_offset_kernel(
    const float* __restrict__ x,    // [N,Cin,H,W]
    const float* __restrict__ pw,   // [Cout,Cin]
    const float* __restrict__ ow,   // [1,Cout]
    const float* __restrict__ ob,   // [1]
    float* __restrict__ y_ws,       // [N,HW,64] NHWC
    float* __restrict__ s_ws)       // [N,HW]
{
    __shared__ float lx[128 * STRX];   // x tile: pixel-major [m][c]
    __shared__ float lw[64 * STRX];    // weights [o][c]
    __shared__ float low[64];          // off_weight

    const int t   = threadIdx.x;
    const int blk = blockIdx.x;
    const int n   = blk >> 9;            // 512 blocks / image
    const int hw0 = (blk & 511) << 7;    // *128

    // stage pw weights (row-major o,c) into padded LDS
    for (int idx = t; idx < 64 * 64; idx += 256) {
        int o = idx >> 6, c = idx & 63;
        lw[o * STRX + c] = pw[idx];
    }
    if (t < 64) low[t] = ow[t];

    // stage x tile transposed: lx[m][c] = x[n, c, hw0+m]
    const float* xb = x + (size_t)n * ((size_t)Cc * HWp) + hw0;
    for (int idx = t; idx < 64 * 128; idx += 256) {
        int c = idx >> 7, m = idx & 127;
        lx[m * STRX + c] = xb[(size_t)c * HWp + m];
    }
    __syncthreads();

    const int lane = t & 31;
    const int wv   = t >> 5;       // wave id 0..7
    const int m0   = wv << 4;      // 16 pixels per wave
    const int lm   = lane & 15;    // N-index / M-index within fragment
    const int hi   = lane >> 4;    // half-wave

    v8f acc0 = {}, acc1 = {}, acc2 = {}, acc3 = {};

    const float* arow = &lx[(m0 + lm) * STRX];
    const float* b0r  = &lw[(0 * 16 + lm) * STRX];
    const float* b1r  = &lw[(1 * 16 + lm) * STRX];
    const float* b2r  = &lw[(2 * 16 + lm) * STRX];
    const float* b3r  = &lw[(3 * 16 + lm) * STRX];

#pragma unroll
    for (int k0 = 0; k0 < 64; k0 += 4) {
        const int ka = k0 + 2 * hi;     // A: lane<16 -> K={k0,k0+1}; lane>=16 -> K={k0+2,k0+3}
        v2f a;  a.x = arow[ka]; a.y = arow[ka + 1];
        v2f b0; b0.x = b0r[ka]; b0.y = b0r[ka + 1];
        v2f b1; b1.x = b1r[ka]; b1.y = b1r[ka + 1];
        v2f b2; b2.x = b2r[ka]; b2.y = b2r[ka + 1];
        v2f b3; b3.x = b3r[ka]; b3.y = b3r[ka + 1];
        acc0 = __builtin_amdgcn_wmma_f32_16x16x4_f32(false, a, false, b0, (short)0, acc0, false, false);
        acc1 = __builtin_amdgcn_wmma_f32_16x16x4_f32(false, a, false, b1, (short)0, acc1, false, false);
        acc2 = __builtin_amdgcn_wmma_f32_16x16x4_f32(false, a, false, b2, (short)0, acc2, false, false);
        acc3 = __builtin_amdgcn_wmma_f32_16x16x4_f32(false, a, false, b3, (short)0, acc3, false, false);
    }

    // store y (NHWC): C/D layout -> lane holds channel lm (+16*nt), vgpr v holds pixel v+8*hi
    const size_t pixbase = (size_t)n * HWp + hw0 + m0;
#pragma unroll
    for (int v = 0; v < 8; ++v) {
        float* yrow = y_ws + (pixbase + v + 8 * hi) * 64 + lm;
        yrow[0]  = acc0[v];
        yrow[16] = acc1[v];
        yrow[32] = acc2[v];
        yrow[48] = acc3[v];
    }

    // offset scalar: s = clip(ow . y + bias, 0, 8); reduce across the 16 lanes of each half
    const float w0 = low[lm], w1 = low[16 + lm], w2 = low[32 + lm], w3 = low[48 + lm];
    const float bias = ob[0];
    float part[8];
#pragma unroll
    for (int v = 0; v < 8; ++v)
        part[v] = w0 * acc0[v] + w1 * acc1[v] + w2 * acc2[v] + w3 * acc3[v];
#pragma unroll
    for (int off = 1; off < 16; off <<= 1) {
#pragma unroll
        for (int v = 0; v < 8; ++v)
            part[v] += __shfl_xor(part[v], off, 32);
    }
    if (lm == 0) {
#pragma unroll
        for (int v = 0; v < 8; ++v) {
            float sv = part[v] + bias;
            sv = fminf(fmaxf(sv, 0.0f), 8.0f);
            s_ws[pixbase + v + 8 * hi] = sv;
        }
    }
}

// ---------------------------------------------------------------------------
// Kernel 2: deformable depthwise 3x3 with bilinear sampling (zero padding).
// Thread = (pixel, 16-channel group). y is NHWC -> aligned float4 gathers.
// ---------------------------------------------------------------------------
__global__ __launch_bounds__(256) void deform_dw_kernel(
    const float* __restrict__ y_ws,   // [N,HW,64]
    const float* __restrict__ s_ws,   // [N,HW]
    const float* __restrict__ dw,     // [64,3,3]
    float* __restrict__ out)          // [N,64,H,W]
{
    __shared__ float dwl[64 * 9];
    const int t = threadIdx.x;
    for (int idx = t; idx < 576; idx += 256) dwl[idx] = dw[idx];
    __syncthreads();

    const int blk = blockIdx.x;
    const int n   = blk >> 10;                      // 1024 blocks / image
    const int hw  = ((blk & 1023) << 6) + (t >> 2); // 64 pixels / block
    const int cg  = t & 3;                          // channel group: cg*16 .. cg*16+15
    const int i   = hw >> 8;
    const int j   = hw & 255;

    const float s = s_ws[(size_t)n * HWp + hw];
    const float* ybase = y_ws + ((size_t)n * HWp) * 64 + cg * 16;

    float acc[16];
#pragma unroll
    for (int c = 0; c < 16; ++c) acc[c] = 0.0f;

#pragma unroll
    for (int k = 0; k < 9; ++k) {
        const int ky = k / 3, kx = k % 3;
        const float dy = (float)(ky - 1), dx = (float)(kx - 1);
        const float py = (float)(i - 1 + ky) + dy * s;
        const float px = (float)(j - 1 + kx) + dx * s;

        const float y0f = floorf(py), x0f = floorf(px);
        const float ly = py - y0f, lxw = px - x0f;
        const int y0 = (int)y0f, x0 = (int)x0f;
        const float wys[2] = {1.0f - ly, ly};
        const float wxs[2] = {1.0f - lxw, lxw};

        float smp[16];
#pragma unroll
        for (int c = 0; c < 16; ++c) smp[c] = 0.0f;

#pragma unroll
        for (int ny = 0; ny < 2; ++ny) {
            const int yy = y0 + ny;
            if (yy < 0 || yy >= Hh) continue;
#pragma unroll
            for (int nx = 0; nx < 2; ++nx) {
                const int xx = x0 + nx;
                if (xx < 0 || xx >= Ww) continue;
                const float wgt = wys[ny] * wxs[nx];
                const float4* p = (const float4*)(ybase + (size_t)(yy * Ww + xx) * 64);
#pragma unroll
                for (int q = 0; q < 4; ++q) {
                    const float4 v = p[q];
                    smp[4 * q + 0] += wgt * v.x;
                    smp[4 * q + 1] += wgt * v.y;
                    smp[4 * q + 2] += wgt * v.z;
                    smp[4 * q + 3] += wgt * v.w;
                }
            }
        }
#pragma unroll
        for (int c = 0; c < 16; ++c)
            acc[c] += smp[c] * dwl[(cg * 16 + c) * 9 + k];
    }

    // write NCHW
    float* ob_ = out + ((size_t)n * 64 + cg * 16) * (size_t)HWp + hw;
#pragma unroll
    for (int c = 0; c < 16; ++c)
        ob_[(size_t)c * HWp] = acc[c];
}

extern "C" void kernel_launch(void* const* d_in, const int* in_sizes, int n_in,
                              void* d_out, int out_size, void* d_ws, size_t ws_size,
                              hipStream_t stream) {
    const float* x  = (const float*)d_in[0];   // [8,64,256,256]
    const float* pw = (const float*)d_in[1];   // [64,64]
    const float* ow = (const float*)d_in[2];   // [1,64]
    const float* ob = (const float*)d_in[3];   // [1]
    const float* dw = (const float*)d_in[4];   // [64,3,3]

    float* y_ws = (float*)d_ws;                              // 8*65536*64 floats = 128 MiB
    float* s_ws = y_ws + (size_t)Nn * HWp * 64;              // 8*65536 floats = 2 MiB

    pw_gemm_offset_kernel<<<4096, 256, 0, stream>>>(x, pw, ow, ob, y_ws, s_ws);
    deform_dw_kernel<<<8192, 256, 0, stream>>>(y_ws, s_ws, dw, (float*)d_out);
}